// CombinedSegmentationLoss_73778948211150
// MI455X (gfx1250) — compile-verified
//
#include <hip/hip_runtime.h>
#include <math.h>

// CombinedSegmentationLoss for MI455X (gfx1250, wave32).
// Strategy:
//  - maxpool_{2d+1} == iterated 3x3 maxpool (ping-pong), -inf padding semantics.
//  - sigmoid monotone => boundary = sigmoid(max3(x)) - sigmoid(min3(x)) straight from logits.
//  - All reductions accumulated through V_WMMA_F32_16X16X4_F32 with all-ones B
//    (D[m,n] += sum_k A[m,k]  =>  sum over D == 16 * sum of fed values).
//  - Deterministic: per-block partials, fixed-order tree/serial reduction, no float atomics.
//  - Native v_exp_f32 / v_rcp_f32 paths instead of libm expf + IEEE divide
//    (bandwidth-bound kernel; ~1-2 ulp elementwise error is negligible in 4M-elem means).

typedef __attribute__((ext_vector_type(2))) float v2f;
typedef __attribute__((ext_vector_type(8))) float v8f;

#define Bn   16
#define Hn   512
#define Wn   512
#define HWn  (Hn*Wn)
#define Ntot (Bn*HWn)
#define ROWS 8
#define TPB  256
#define BLOCKS_PER_IMG (Hn/ROWS)       // 64
#define NBLK (Bn*BLOCKS_PER_IMG)       // 1024

// workspace float layout
//  [0,N)        predB          [N,2N)  gtB
//  [2N,3N) bufA  [3N,4N) bufB  [4N,5N) bufC  [5N,6N) bufD
//  [6N, ...)    partials
#define PART_BCE 0
#define PART_P   1024
#define PART_T   2048
#define PART_PT  3072
#define PART_HD  4096                  // 20 arrays of 1024
#define PART_TOTAL (PART_HD + 20*1024)

__device__ __forceinline__ v8f wmma_sum(v2f a, v8f c) {
  const v2f ones = {1.0f, 1.0f};
  // D = A(16x4) * ones(4x16) + C ; every A element lands in 16 of 256 D slots.
  return __builtin_amdgcn_wmma_f32_16x16x4_f32(false, a, false, ones, (short)0, c,
                                               false, false);
}

// full-wave (32-lane) sum of a v8f accumulator; every lane returns the total
__device__ __forceinline__ float wave_total(v8f c) {
  float s = 0.0f;
#pragma unroll
  for (int r = 0; r < 8; ++r) s += c[r];
#pragma unroll
  for (int off = 16; off >= 1; off >>= 1) s += __shfl_xor(s, off, 32);
  return s;
}

__device__ __forceinline__ float sigmoidf_(float x) {
  // v_exp_f32 + v_rcp_f32; monotone, ~2 ulp
  return __builtin_amdgcn_rcpf(1.0f + __expf(-x));
}

// ---------------------------------------------------------------------------
// Kernel 1: fused BCE / per-batch p,t,pt sums / soft boundaries.
// One block = 8 rows of one image. LDS tile: 10 rows x 512 cols (logits+target).
// ---------------------------------------------------------------------------
__global__ __launch_bounds__(TPB) void k_boundary_reduce(
    const float* __restrict__ logits, const int* __restrict__ target,
    float* __restrict__ predB, float* __restrict__ gtB,
    float* __restrict__ part)
{
  __shared__ float sl[(ROWS + 2) * Wn];
  __shared__ float st[(ROWS + 2) * Wn];
  __shared__ float red[4][TPB / 32];

  const int tid = threadIdx.x;
  const int blk = blockIdx.x;
  const int b   = blk / BLOCKS_PER_IMG;
  const int y0  = (blk % BLOCKS_PER_IMG) * ROWS;
  const size_t imgOff = (size_t)b * HWn;
  const float* limg = logits + imgOff;
  const int*   timg = target + imgOff;

#pragma unroll
  for (int j = 0; j < (ROWS + 2) * Wn / TPB; ++j) {
    int idx = tid + j * TPB;
    int r = idx / Wn, c = idx - r * Wn;
    int gy = y0 - 1 + r;
    float lv = 0.0f, tv = 0.0f;
    if (gy >= 0 && gy < Hn) {
      lv = limg[gy * Wn + c];
      tv = (float)timg[gy * Wn + c];
    }
    sl[idx] = lv;
    st[idx] = tv;
  }
  __syncthreads();

  v8f aB = {}; v8f aP = {}; v8f aT = {}; v8f aPT = {};

  for (int it = 0; it < ROWS; ++it) {
    const int pr = it;
    const int gy = y0 + pr;
    const int x0 = tid * 2;
    float bcev[2], pv[2], tcv[2], ptv[2];
#pragma unroll
    for (int s2 = 0; s2 < 2; ++s2) {
      const int x = x0 + s2;
      const float xc = sl[(pr + 1) * Wn + x];
      const float tc = st[(pr + 1) * Wn + x];
      // stable BCE: max(x,0) - x*t + log1p(exp(-|x|)); exp via native path,
      // log1p argument is in (0,1] so no cancellation risk.
      const float bce = fmaxf(xc, 0.0f) - xc * tc + log1pf(__expf(-fabsf(xc)));
      const float p = sigmoidf_(xc);

      float mx = xc, mn = xc, tmx = tc, tmn = tc;
#pragma unroll
      for (int dy = -1; dy <= 1; ++dy) {
        const int gyy = gy + dy;
        if (gyy < 0 || gyy >= Hn) continue;   // -inf / +inf padding semantics
        const int lr = pr + 1 + dy;
#pragma unroll
        for (int dx = -1; dx <= 1; ++dx) {
          const int xx = x + dx;
          if (xx < 0 || xx >= Wn) continue;
          const float v = sl[lr * Wn + xx];
          const float w = st[lr * Wn + xx];
          mx = fmaxf(mx, v); mn = fminf(mn, v);
          tmx = fmaxf(tmx, w); tmn = fminf(tmn, w);
        }
      }
      const float pb = sigmoidf_(mx) - sigmoidf_(mn);   // >= 0 by monotonicity
      const float gb = tmx - tmn;                       // >= 0
      const size_t gidx = imgOff + (size_t)gy * Wn + x;
      predB[gidx] = pb;
      gtB[gidx]   = gb;
      bcev[s2] = bce; pv[s2] = p; tcv[s2] = tc; ptv[s2] = p * tc;
    }
    v2f a;
    a.x = bcev[0]; a.y = bcev[1];  aB  = wmma_sum(a, aB);
    a.x = pv[0];   a.y = pv[1];    aP  = wmma_sum(a, aP);
    a.x = tcv[0];  a.y = tcv[1];   aT  = wmma_sum(a, aT);
    a.x = ptv[0];  a.y = ptv[1];   aPT = wmma_sum(a, aPT);
  }

  const float sB = wave_total(aB), sP = wave_total(aP);
  const float sT = wave_total(aT), sPT = wave_total(aPT);
  const int wave = tid >> 5, lane = tid & 31;
  if (lane == 0) {
    red[0][wave] = sB; red[1][wave] = sP; red[2][wave] = sT; red[3][wave] = sPT;
  }
  __syncthreads();
  if (tid < 4) {
    float s = 0.0f;
#pragma unroll
    for (int w = 0; w < TPB / 32; ++w) s += red[tid][w];
    s *= (1.0f / 16.0f);   // undo 16x replication of the WMMA ones-reduction
    float* dsts[4] = {part + PART_BCE, part + PART_P, part + PART_T, part + PART_PT};
    dsts[tid][blk] = s;
  }
}

// ---------------------------------------------------------------------------
// Kernel 2 (x10): one 3x3 max-pool step for both boundary chains + abs-diff
// partial sums (WMMA-accumulated).
// ---------------------------------------------------------------------------
__global__ __launch_bounds__(TPB) void k_hd_step(
    const float* __restrict__ curP, const float* __restrict__ curG,
    const float* __restrict__ predB, const float* __restrict__ gtB,
    float* __restrict__ nextP, float* __restrict__ nextG,
    float* __restrict__ part1, float* __restrict__ part2)
{
  __shared__ float sp[(ROWS + 2) * (Wn + 2)];
  __shared__ float sg[(ROWS + 2) * (Wn + 2)];
  __shared__ float red[2][TPB / 32];

  const int tid = threadIdx.x;
  const int blk = blockIdx.x;
  const int b   = blk / BLOCKS_PER_IMG;
  const int y0  = (blk % BLOCKS_PER_IMG) * ROWS;
  const size_t imgOff = (size_t)b * HWn;

  // prefetch streamed reference boundaries for this tile (global_prefetch_b8)
  __builtin_prefetch(gtB + imgOff + (size_t)y0 * Wn, 0, 0);
  __builtin_prefetch(predB + imgOff + (size_t)y0 * Wn, 0, 0);

  const float NEG = -3.0e38f;
  for (int idx = tid; idx < (ROWS + 2) * (Wn + 2); idx += TPB) {
    const int r = idx / (Wn + 2);
    const int c = idx - r * (Wn + 2);
    const int gy = y0 - 1 + r;
    const int gx = c - 1;
    float pv = NEG, gv = NEG;
    if (gy >= 0 && gy < Hn && gx >= 0 && gx < Wn) {
      pv = curP[imgOff + (size_t)gy * Wn + gx];
      gv = curG[imgOff + (size_t)gy * Wn + gx];
    }
    sp[idx] = pv;
    sg[idx] = gv;
  }
  __syncthreads();

  v8f a1 = {}; v8f a2 = {};

  for (int it = 0; it < ROWS; ++it) {
    const int pr = it;
    const int gy = y0 + pr;
    const int x0 = tid * 2;
    float d1v[2], d2v[2];
#pragma unroll
    for (int s2 = 0; s2 < 2; ++s2) {
      const int x = x0 + s2;      // LDS center col = x+1; window cols x..x+2
      float pd = sp[pr * (Wn + 2) + x];
      float gd = sg[pr * (Wn + 2) + x];
#pragma unroll
      for (int dy = 0; dy < 3; ++dy) {
#pragma unroll
        for (int dx = 0; dx < 3; ++dx) {
          pd = fmaxf(pd, sp[(pr + dy) * (Wn + 2) + x + dx]);
          gd = fmaxf(gd, sg[(pr + dy) * (Wn + 2) + x + dx]);
        }
      }
      const size_t gidx = imgOff + (size_t)gy * Wn + x;
      nextP[gidx] = pd;
      nextG[gidx] = gd;
      d1v[s2] = fabsf(pd - gtB[gidx]);
      d2v[s2] = fabsf(gd - predB[gidx]);
    }
    v2f a;
    a.x = d1v[0]; a.y = d1v[1];  a1 = wmma_sum(a, a1);
    a.x = d2v[0]; a.y = d2v[1];  a2 = wmma_sum(a, a2);
  }

  const float s1 = wave_total(a1), s2t = wave_total(a2);
  const int wave = tid >> 5, lane = tid & 31;
  if (lane == 0) { red[0][wave] = s1; red[1][wave] = s2t; }
  __syncthreads();
  if (tid < 2) {
    float s = 0.0f;
#pragma unroll
    for (int w = 0; w < TPB / 32; ++w) s += red[tid][w];
    s *= (1.0f / 16.0f);
    (tid == 0 ? part1 : part2)[blk] = s;
  }
}

// ---------------------------------------------------------------------------
// Finalize: deterministic fixed-order reduction of all partials + loss formula.
// ---------------------------------------------------------------------------
__global__ __launch_bounds__(64) void k_finalize(const float* __restrict__ part,
                                                 float* __restrict__ out)
{
  __shared__ float shD[16], shF[16], shH[20], shB[16];
  const int tid = threadIdx.x;
  const float invN = 1.0f / (float)Ntot;

  if (tid < 16) {                       // per-batch dice + focal-tversky
    float P = 0.0f, T = 0.0f, PT = 0.0f;
    for (int i = 0; i < BLOCKS_PER_IMG; ++i) {
      P  += part[PART_P  + tid * BLOCKS_PER_IMG + i];
      T  += part[PART_T  + tid * BLOCKS_PER_IMG + i];
      PT += part[PART_PT + tid * BLOCKS_PER_IMG + i];
    }
    const float dice = (2.0f * PT + 1e-6f) / (P + T + 1e-6f + 1e-7f);
    const float FPv = P - PT, FNv = T - PT;
    const float tv = (PT + 1e-6f) /
                     (PT + 0.7f * FPv + 0.3f * FNv + 1e-6f + 1e-7f);
    shD[tid] = 1.0f - dice;
    shF[tid] = powf(fmaxf(1.0f - tv, 0.0f), 0.75f);
  } else if (tid < 36) {                // 20 hausdorff sums
    const int q = tid - 16;
    float s = 0.0f;
    for (int i = 0; i < NBLK; ++i) s += part[PART_HD + q * NBLK + i];
    shH[q] = s;
  } else if (tid < 52) {                // bce in 16 fixed chunks
    const int q = tid - 36;
    float s = 0.0f;
    for (int i = 0; i < BLOCKS_PER_IMG; ++i)
      s += part[PART_BCE + q * BLOCKS_PER_IMG + i];
    shB[q] = s;
  }
  __syncthreads();
  if (tid == 0) {
    float dsum = 0.0f, fsum = 0.0f, bsum = 0.0f;
    for (int i = 0; i < 16; ++i) { dsum += shD[i]; fsum += shF[i]; bsum += shB[i]; }
    float hd = 0.0f, wsum = 0.0f;
    for (int d = 1; d <= 10; ++d) {
      const float w = (float)d * 0.1f;
      hd += w * (shH[2 * (d - 1)] + shH[2 * (d - 1) + 1]) * invN;
      wsum += w;
    }
    hd /= (wsum + 1e-8f);
    out[0] = bsum * invN + dsum * (1.0f / 16.0f) + fsum * (1.0f / 16.0f) + 0.1f * hd;
  }
}

// ---------------------------------------------------------------------------
extern "C" void kernel_launch(void* const* d_in, const int* in_sizes, int n_in,
                              void* d_out, int out_size, void* d_ws, size_t ws_size,
                              hipStream_t stream)
{
  (void)in_sizes; (void)n_in; (void)out_size; (void)ws_size;
  const float* logits = (const float*)d_in[0];
  const int*   target = (const int*)d_in[1];
  float* ws = (float*)d_ws;

  float* predB = ws;
  float* gtB   = ws + (size_t)Ntot;
  float* bufA  = ws + 2 * (size_t)Ntot;
  float* bufB  = ws + 3 * (size_t)Ntot;
  float* bufC  = ws + 4 * (size_t)Ntot;
  float* bufD  = ws + 5 * (size_t)Ntot;
  float* part  = ws + 6 * (size_t)Ntot;   // PART_TOTAL floats, fully rewritten each call

  k_boundary_reduce<<<NBLK, TPB, 0, stream>>>(logits, target, predB, gtB, part);

  const float* cP = predB; const float* cG = gtB;
  float* oP = bufA; float* oG = bufC;
  for (int d = 1; d <= 10; ++d) {
    float* p1 = part + PART_HD + (size_t)(2 * (d - 1)) * NBLK;
    float* p2 = p1 + NBLK;
    k_hd_step<<<NBLK, TPB, 0, stream>>>(cP, cG, predB, gtB, oP, oG, p1, p2);
    cP = oP; cG = oG;
    if (d & 1) { oP = bufB; oG = bufD; } else { oP = bufA; oG = bufC; }
  }

  k_finalize<<<1, 64, 0, stream>>>(part, (float*)d_out);
}